// NodeScoringNN_68040871903279
// MI455X (gfx1250) — compile-verified
//
#include <hip/hip_runtime.h>
#include <hip/hip_bf16.h>

typedef __attribute__((ext_vector_type(2))) float v2f;
typedef __attribute__((ext_vector_type(8))) float v8f;

#define D_DIM 512
#define NCLUST 64

// ---------------------------------------------------------------------------
// init: zero control structures (sizes, prefix/need/eqcnt, histograms)
// ---------------------------------------------------------------------------
__global__ void initK(unsigned* p, int count) {
    int i = blockIdx.x * blockDim.x + threadIdx.x;
    int stride = gridDim.x * blockDim.x;
    for (; i < count; i += stride) p[i] = 0u;
}

// ---------------------------------------------------------------------------
// fold the linear network: vv[d] = sum_h W2[h]*W1[h,d];  s0 = b1.W2 + b2
// one block of 512 threads
// ---------------------------------------------------------------------------
__global__ void foldK(const float* __restrict__ W1, const float* __restrict__ b1,
                      const float* __restrict__ W2, const float* __restrict__ b2,
                      float* __restrict__ vv, float* __restrict__ s0) {
    __shared__ float red[D_DIM];
    int d = threadIdx.x;
    float acc = 0.f;
    for (int h = 0; h < D_DIM; ++h)
        acc += W2[h] * W1[h * D_DIM + d];
    vv[d] = acc;
    red[d] = b1[d] * W2[d];
    __syncthreads();
    for (int s = D_DIM / 2; s > 0; s >>= 1) {
        if (d < s) red[d] += red[d + s];
        __syncthreads();
    }
    if (d == 0) *s0 = red[0] + b2[0];
}

// ---------------------------------------------------------------------------
// scores via V_WMMA_F32_16X16X4_F32. One wave per 16-row tile.
// A = 16x4 tile of x, B = 4x16 with all 16 columns = v  => every D column is
// the fp32 dot product x_row . v. Emit monotone order key of the logit.
// ---------------------------------------------------------------------------
__global__ void scoreK(const float* __restrict__ x, const float* __restrict__ vv,
                       const float* __restrict__ s0p, unsigned* __restrict__ keys,
                       int n) {
    __shared__ float vs[D_DIM];
    for (int i = threadIdx.x; i < D_DIM; i += blockDim.x) vs[i] = vv[i];
    __syncthreads();

    int ntiles = (n + 15) >> 4;
    int tile = blockIdx.x * (blockDim.x >> 5) + (threadIdx.x >> 5);
    if (tile >= ntiles) return;

    int lane = threadIdx.x & 31;
    int m    = lane & 15;
    int hi   = lane >> 4;          // 0: K lo pair, 1: K hi pair (ISA 32-bit A/B layout)
    int off2 = hi * 2;

    int row = tile * 16 + m;
    if (row >= n) row = n - 1;     // clamp (n is a multiple of 16 anyway)
    const float* xrow = x + (size_t)row * D_DIM + off2;
    const float* vrow = vs + off2;

    v8f c = {};
#pragma unroll 8
    for (int k0 = 0; k0 < D_DIM; k0 += 4) {
        v2f a; a.x = xrow[k0]; a.y = xrow[k0 + 1];
        v2f b; b.x = vrow[k0]; b.y = vrow[k0 + 1];
        // D = A(16x4) * B(4x16) + C   (fp32 WMMA)
        c = __builtin_amdgcn_wmma_f32_16x16x4_f32(
                false, a, false, b, (short)0, c, false, false);
    }

    float s0 = *s0p;
    // C/D layout: lanes 0-15 -> M = vgpr (N=lane); lanes 16-31 -> M = 8+vgpr.
    if (m == 0) {
        int base = tile * 16 + hi * 8;
#pragma unroll
        for (int i = 0; i < 8; ++i) {
            int r = base + i;
            if (r < n) {
                float logit = c[i] + s0;          // sigmoid is monotone: rank by logit
                unsigned bits = __float_as_uint(logit);
                unsigned key = ((int)bits < 0) ? ~bits : (bits | 0x80000000u);
                keys[r] = key;
            }
        }
    }
}

// ---------------------------------------------------------------------------
// cluster sizes
// ---------------------------------------------------------------------------
__global__ void sizesK(const int* __restrict__ c, unsigned* __restrict__ sizes, int n) {
    __shared__ unsigned h[NCLUST];
    if (threadIdx.x < NCLUST) h[threadIdx.x] = 0u;
    __syncthreads();
    int i = blockIdx.x * blockDim.x + threadIdx.x;
    int stride = gridDim.x * blockDim.x;
    for (; i < n; i += stride) atomicAdd(&h[c[i] & (NCLUST - 1)], 1u);
    __syncthreads();
    if (threadIdx.x < NCLUST && h[threadIdx.x])
        atomicAdd(&sizes[threadIdx.x], h[threadIdx.x]);
}

// ---------------------------------------------------------------------------
// quota drain (sequential, matches reference scan); init radix-select state.
// prefix/need index 64 is the "global leftover" selector.
// ---------------------------------------------------------------------------
__global__ void quotaK(const unsigned* __restrict__ sizes, const int* __restrict__ budget_p,
                       int n, unsigned* __restrict__ prefix, unsigned* __restrict__ need) {
    __shared__ int want[NCLUST];
    int cc = threadIdx.x;
    int budget = *budget_p;
    if (cc < NCLUST) {
        float w = (float)budget * (float)sizes[cc] / (float)n;
        want[cc] = (int)rintf(w);   // round half to even, like jnp.round
    }
    __syncthreads();
    if (cc == 0) {
        int rem = budget;
        for (int j = 0; j < NCLUST; ++j) {
            int q = want[j] < rem ? want[j] : rem;
            rem -= q;
            need[j]   = (unsigned)q;
            prefix[j] = (q == 0) ? 0xFFFFFFFFu : 0u;
        }
        need[NCLUST]   = (unsigned)rem;
        prefix[NCLUST] = (rem == 0) ? 0xFFFFFFFFu : 0u;
    }
}

// ---------------------------------------------------------------------------
// per-cluster radix select, one 8-bit digit per pass (shift = 24,16,8,0)
// ---------------------------------------------------------------------------
__global__ void histClusterK(const unsigned* __restrict__ keys, const int* __restrict__ c,
                             const unsigned* __restrict__ prefix,
                             const unsigned* __restrict__ need,
                             unsigned* __restrict__ hist, int n, int shift) {
    unsigned maskHi = (shift >= 24) ? 0u : (0xFFFFFFFFu << (shift + 8));
    int i = blockIdx.x * blockDim.x + threadIdx.x;
    int stride = gridDim.x * blockDim.x;
    for (; i < n; i += stride) {
        int cc = c[i] & (NCLUST - 1);
        if (need[cc] == 0u) continue;
        unsigned k = keys[i];
        if ((k & maskHi) == (prefix[cc] & maskHi))
            atomicAdd(&hist[cc * 256 + ((k >> shift) & 255u)], 1u);
    }
}

__global__ void scanClusterK(unsigned* __restrict__ hist, unsigned* __restrict__ prefix,
                             unsigned* __restrict__ need, int shift) {
    int cc = threadIdx.x;
    if (cc >= NCLUST) return;
    unsigned nd = need[cc];
    unsigned* h = hist + cc * 256;
    if (nd > 0u) {
        unsigned cum = 0u, above = 0u;
        int chosen = 0;
        for (int b = 255; b >= 0; --b) {
            unsigned hb = h[b];
            if (cum + hb >= nd) { chosen = b; above = cum; break; }
            cum += hb;
        }
        prefix[cc] |= ((unsigned)chosen) << shift;
        need[cc] = nd - above;     // how many still needed within chosen bucket
    }
    for (int b = 0; b < 256; ++b) h[b] = 0u;   // ready for next pass
}

// sel1: per-cluster top-quota -> write 0/1 directly into out
__global__ void sel1K(const unsigned* __restrict__ keys, const int* __restrict__ c,
                      const unsigned* __restrict__ prefix, const unsigned* __restrict__ need,
                      unsigned* __restrict__ eqcnt, float* __restrict__ out, int n) {
    int i = blockIdx.x * blockDim.x + threadIdx.x;
    int stride = gridDim.x * blockDim.x;
    for (; i < n; i += stride) {
        int cc = c[i] & (NCLUST - 1);
        unsigned T = prefix[cc];
        unsigned k = keys[i];
        float sel = 0.f;
        if (k > T) sel = 1.f;
        else if (k == T) {
            unsigned pos = atomicAdd(&eqcnt[cc], 1u);
            if (pos < need[cc]) sel = 1.f;
        }
        out[i] = sel;
    }
}

// leftover budget: global radix select over unselected nodes
__global__ void histGlobalK(const unsigned* __restrict__ keys, const float* __restrict__ out,
                            const unsigned* __restrict__ prefix, const unsigned* __restrict__ need,
                            unsigned* __restrict__ histG, int n, int shift) {
    if (need[NCLUST] == 0u) return;
    unsigned maskHi = (shift >= 24) ? 0u : (0xFFFFFFFFu << (shift + 8));
    unsigned pfx = prefix[NCLUST];
    int i = blockIdx.x * blockDim.x + threadIdx.x;
    int stride = gridDim.x * blockDim.x;
    for (; i < n; i += stride) {
        if (out[i] != 0.f) continue;
        unsigned k = keys[i];
        if ((k & maskHi) == (pfx & maskHi))
            atomicAdd(&histG[(k >> shift) & 255u], 1u);
    }
}

__global__ void scanGlobalK(unsigned* __restrict__ histG, unsigned* __restrict__ prefix,
                            unsigned* __restrict__ need, int shift) {
    if (threadIdx.x != 0 || blockIdx.x != 0) return;
    unsigned nd = need[NCLUST];
    if (nd > 0u) {
        unsigned cum = 0u, above = 0u;
        int chosen = 0;
        for (int b = 255; b >= 0; --b) {
            unsigned hb = histG[b];
            if (cum + hb >= nd) { chosen = b; above = cum; break; }
            cum += hb;
        }
        prefix[NCLUST] |= ((unsigned)chosen) << shift;
        need[NCLUST] = nd - above;
    }
    for (int b = 0; b < 256; ++b) histG[b] = 0u;
}

__global__ void sel2K(const unsigned* __restrict__ keys, const unsigned* __restrict__ prefix,
                      const unsigned* __restrict__ need, unsigned* __restrict__ eqcnt,
                      float* __restrict__ out, int n) {
    int i = blockIdx.x * blockDim.x + threadIdx.x;
    int stride = gridDim.x * blockDim.x;
    unsigned T = prefix[NCLUST];
    for (; i < n; i += stride) {
        if (out[i] != 0.f) continue;
        unsigned k = keys[i];
        if (k > T) out[i] = 1.f;
        else if (k == T) {
            unsigned pos = atomicAdd(&eqcnt[NCLUST], 1u);
            if (pos < need[NCLUST]) out[i] = 1.f;
        }
    }
}

// ---------------------------------------------------------------------------
extern "C" void kernel_launch(void* const* d_in, const int* in_sizes, int n_in,
                              void* d_out, int out_size, void* d_ws, size_t ws_size,
                              hipStream_t stream) {
    // inputs (setup_inputs order): x, c, k, W1, b1, W2, b2
    const float* x  = (const float*)d_in[0];
    const int*   c  = (const int*)  d_in[1];
    const int*   kB = (const int*)  d_in[2];   // budget (device scalar)
    const float* W1 = (const float*)d_in[3];
    const float* b1 = (const float*)d_in[4];
    const float* W2 = (const float*)d_in[5];
    const float* b2 = (const float*)d_in[6];
    float* out = (float*)d_out;
    int n = in_sizes[1];                       // number of nodes

    // workspace layout (32-bit words)
    unsigned* wsu   = (unsigned*)d_ws;
    float*    vv    = (float*)wsu;             // 512
    float*    s0    = (float*)(wsu + D_DIM);   // 1
    unsigned* keys  = wsu + 1024;              // n
    unsigned* sizes = keys + n;                // 64
    unsigned* prefix= sizes + NCLUST;          // 65
    unsigned* need  = prefix + (NCLUST + 1);   // 65
    unsigned* eqcnt = need + (NCLUST + 1);     // 65
    unsigned* hist  = eqcnt + (NCLUST + 1);    // 64*256 cluster + 256 global
    unsigned* histG = hist + NCLUST * 256;
    int ctlWords = NCLUST + 3 * (NCLUST + 1) + NCLUST * 256 + 256;

    // 1. zero control state
    initK<<<64, 256, 0, stream>>>(sizes, ctlWords);
    // 2. fold network into a single vector + scalar
    foldK<<<1, D_DIM, 0, stream>>>(W1, b1, W2, b2, vv, s0);
    // 3. WMMA mat-vec: order keys for all nodes (bandwidth-bound pass over x)
    {
        int ntiles = (n + 15) / 16;
        int blocks = (ntiles + 7) / 8;         // 8 waves of 32 per block
        scoreK<<<blocks, 256, 0, stream>>>(x, vv, s0, keys, n);
    }
    // 4. cluster sizes + quota drain
    sizesK<<<256, 256, 0, stream>>>(c, sizes, n);
    quotaK<<<1, NCLUST, 0, stream>>>(sizes, kB, n, prefix, need);
    // 5. per-cluster radix select (4 x 8-bit digits, high to low)
    for (int shift = 24; shift >= 0; shift -= 8) {
        histClusterK<<<256, 256, 0, stream>>>(keys, c, prefix, need, hist, n, shift);
        scanClusterK<<<1, NCLUST, 0, stream>>>(hist, prefix, need, shift);
    }
    // 6. per-cluster top-quota selection -> out
    sel1K<<<(n + 255) / 256, 256, 0, stream>>>(keys, c, prefix, need, eqcnt, out, n);
    // 7. leftover budget: global radix select over unselected
    for (int shift = 24; shift >= 0; shift -= 8) {
        histGlobalK<<<256, 256, 0, stream>>>(keys, out, prefix, need, histG, n, shift);
        scanGlobalK<<<1, 64, 0, stream>>>(histG, prefix, need, shift);
    }
    sel2K<<<(n + 255) / 256, 256, 0, stream>>>(keys, prefix, need, eqcnt, out, n);
}